// RoiPooling_26250840113278
// MI455X (gfx1250) — compile-verified
//
#include <hip/hip_runtime.h>

// RoI crop: out[n, b, i, j, c] = in[b, clamp(x1[n])+i, clamp(y1[n])+j, c]
// in : [8, 64, 64, 256] f32 (33.5 MB, L2-resident on 192MB L2)
// out: [N, 8, 7, 7, 256] f32 (~401 MB) -> pure HBM-write-bound copy (~17us floor).
// One 7KB contiguous row-copy per block; 448 lanes x float4 = 7168 B.
// CDNA5 path: async global->LDS->global DMA (ASYNCcnt), no VGPR round-trip.

#define ANCHOR   7
#define FSIZE    64
#define NCH      256
#define CHUNK_F4 448           // 7*256/4 float4 per (n,b,i) row-slab
#define RPB      56            // 8*7 slabs per box

__global__ __launch_bounds__(CHUNK_F4) void roi_crop_async(
    const float* __restrict__ fmap,   // [8,64,64,256]
    const int*   __restrict__ boxes,  // [N,4] x1,y1,x2,y2
    float*       __restrict__ out)    // [N,8,7,7,256]
{
    __shared__ float4 buf[CHUNK_F4];  // 7168 B, one private 16B slot per lane

    const unsigned t   = threadIdx.x;        // 0..447 -> j*64 + c/4
    const unsigned blk = blockIdx.x;         // n*56 + r
    const unsigned n   = blk / RPB;
    const unsigned r   = blk % RPB;          // b*7 + i
    const unsigned b   = r / ANCHOR;
    const unsigned i   = r % ANCHOR;

    int x1 = boxes[n * 4 + 0];
    int y1 = boxes[n * 4 + 1];
    x1 = min(max(x1, 0), FSIZE - ANCHOR);
    y1 = min(max(y1, 0), FSIZE - ANCHOR);

    // src: start of in[b, x1+i, y1, 0], then t*4 floats into the 1792-float row
    const size_t src = ((((size_t)b * FSIZE + (size_t)(x1 + (int)i)) * FSIZE)
                        + (size_t)y1) * NCH + (size_t)t * 4u;
    // dst: out flat = n*100352 + r*1792 + t*4
    const size_t dst = (size_t)blk * (ANCHOR * NCH) + (size_t)t * 4u;

#if defined(__AMDGCN__)
    // ---- device path: async DMA through LDS (gfx1250) ----
  #if __has_builtin(__builtin_amdgcn_global_load_async_to_lds_b128) && \
      __has_builtin(__builtin_amdgcn_global_store_async_from_lds_b128)
    // Builtins type-check as (v4i AS1*, v4i AS3*, imm offset, imm cpol).
    typedef int v4i __attribute__((ext_vector_type(4)));
    typedef __attribute__((address_space(1))) v4i* gas_v4i;
    typedef __attribute__((address_space(3))) v4i* las_v4i;

    gas_v4i gs = (gas_v4i)(fmap + src);
    gas_v4i gd = (gas_v4i)(out + dst);
    las_v4i ls = (las_v4i)&buf[t];

    __builtin_amdgcn_global_load_async_to_lds_b128(gs, ls, 0, 0);
    #if __has_builtin(__builtin_amdgcn_s_wait_asynccnt)
    __builtin_amdgcn_s_wait_asynccnt(0);
    #else
    asm volatile("s_wait_asynccnt 0" ::: "memory");
    #endif
    __builtin_amdgcn_global_store_async_from_lds_b128(gd, ls, 0, 0);
    // store drains at S_ENDPGM (implicit wait-idle)
  #else
    // inline-asm fallback per CDNA5 ISA / CDNA5 ISA:
    //   load : vdst = LDS byte-addr VGPR, vaddr = 64-bit global addr pair
    //   store: vaddr first, then LDS byte-addr VGPR
    const float4* s = (const float4*)(fmap + src);
    float4*       d = (float4*)(out + dst);
    unsigned ldsa = (unsigned)(unsigned long long)&buf[t]; // low 32b of generic = LDS offset
    asm volatile("global_load_async_to_lds_b128 %0, %1, off"
                 :: "v"(ldsa), "v"(s) : "memory");
    asm volatile("s_wait_asynccnt 0" ::: "memory");
    asm volatile("global_store_async_from_lds_b128 %0, %1, off"
                 :: "v"(d), "v"(ldsa) : "memory");
  #endif
#else
    // host-parse placeholder (never executed; device pass uses async path)
    float4*       d = (float4*)(out + dst);
    const float4* s = (const float4*)(fmap + src);
    *d = *s;
    (void)buf;
#endif
}

extern "C" void kernel_launch(void* const* d_in, const int* in_sizes, int n_in,
                              void* d_out, int out_size, void* d_ws, size_t ws_size,
                              hipStream_t stream) {
    (void)n_in; (void)out_size; (void)d_ws; (void)ws_size;
    const float* fmap  = (const float*)d_in[0];
    const int*   boxes = (const int*)d_in[1];
    float*       out   = (float*)d_out;

    const int nboxes = in_sizes[1] / 4;      // [N,4] int32 boxes
    dim3 grid((unsigned)(nboxes * RPB));     // 56,000 blocks of one 7KB copy each
    dim3 block(CHUNK_F4);                    // 448 threads = 14 waves (wave32)
    roi_crop_async<<<grid, block, 0, stream>>>(fmap, boxes, out);
}